// _SparseCodebook_7765300871586
// MI455X (gfx1250) — compile-verified
//
#include <hip/hip_runtime.h>

typedef __attribute__((ext_vector_type(2))) float v2f;
typedef __attribute__((ext_vector_type(4))) float v4f;
typedef __attribute__((ext_vector_type(8))) float v8f;

// One wave32 handles 16 samples. WMMA 16x16x4 F32 with B = ones performs the
// D-dimension reduction on the matrix pipe:
//   D[m][n] = sum_k |codes[m][k] - cent[m][k]| + C[m][n]   (all columns equal)
// A-matrix layout (32-bit, 16x4): lanes 0-15 hold K=0,1 ; lanes 16-31 hold K=2,3.
// We exploit sum-order invariance: lane-half 0 covers dims [0,128),
// lane-half 1 covers dims [128,256)  ->  contiguous per-lane streams -> b128 loads.
__global__ __launch_bounds__(256) void _SparseCodebook_kernel(
    const float* __restrict__ codes,      // [B, 256]
    const int*   __restrict__ pred_class, // [B]
    const float* __restrict__ centroids,  // [1000, 4, 256]
    float*       __restrict__ out)        // [B]
{
    const int tid   = threadIdx.x;
    const int lane  = tid & 31;
    const int wave  = tid >> 5;
    const int group = blockIdx.x * 8 + wave;   // 16-sample group index
    const int row   = lane & 15;               // sample within group (matrix row M)
    const int half  = lane >> 4;               // 0: dims [0,128)  1: dims [128,256)

    const long sample = (long)group * 16 + row;
    const int  cls    = pred_class[sample];

    const float* codep = codes     + sample * 256       + half * 128;
    const float* centp = centroids + (long)cls * 1024   + half * 128;

    const v2f ones = (v2f){1.0f, 1.0f};        // B matrix: 4x16 of 1.0 (2 VGPRs)
    v8f acc0 = {}; v8f acc1 = {}; v8f acc2 = {}; v8f acc3 = {};

#pragma unroll 2
    for (int s = 0; s < 32; ++s) {
        const int off = s * 4;                 // 4 dims per lane per iteration
        const v4f a  = *(const v4f*)(codep + off);
        const v4f c0 = *(const v4f*)(centp + 0 * 256 + off);
        const v4f c1 = *(const v4f*)(centp + 1 * 256 + off);
        const v4f c2 = *(const v4f*)(centp + 2 * 256 + off);
        const v4f c3 = *(const v4f*)(centp + 3 * 256 + off);

        v2f d;
        // centroid 0
        d = (v2f){__builtin_fabsf(a.x - c0.x), __builtin_fabsf(a.y - c0.y)};
        acc0 = __builtin_amdgcn_wmma_f32_16x16x4_f32(false, d, false, ones, (short)0, acc0, false, false);
        d = (v2f){__builtin_fabsf(a.z - c0.z), __builtin_fabsf(a.w - c0.w)};
        acc0 = __builtin_amdgcn_wmma_f32_16x16x4_f32(false, d, false, ones, (short)0, acc0, false, false);
        // centroid 1
        d = (v2f){__builtin_fabsf(a.x - c1.x), __builtin_fabsf(a.y - c1.y)};
        acc1 = __builtin_amdgcn_wmma_f32_16x16x4_f32(false, d, false, ones, (short)0, acc1, false, false);
        d = (v2f){__builtin_fabsf(a.z - c1.z), __builtin_fabsf(a.w - c1.w)};
        acc1 = __builtin_amdgcn_wmma_f32_16x16x4_f32(false, d, false, ones, (short)0, acc1, false, false);
        // centroid 2
        d = (v2f){__builtin_fabsf(a.x - c2.x), __builtin_fabsf(a.y - c2.y)};
        acc2 = __builtin_amdgcn_wmma_f32_16x16x4_f32(false, d, false, ones, (short)0, acc2, false, false);
        d = (v2f){__builtin_fabsf(a.z - c2.z), __builtin_fabsf(a.w - c2.w)};
        acc2 = __builtin_amdgcn_wmma_f32_16x16x4_f32(false, d, false, ones, (short)0, acc2, false, false);
        // centroid 3
        d = (v2f){__builtin_fabsf(a.x - c3.x), __builtin_fabsf(a.y - c3.y)};
        acc3 = __builtin_amdgcn_wmma_f32_16x16x4_f32(false, d, false, ones, (short)0, acc3, false, false);
        d = (v2f){__builtin_fabsf(a.z - c3.z), __builtin_fabsf(a.w - c3.w)};
        acc3 = __builtin_amdgcn_wmma_f32_16x16x4_f32(false, d, false, ones, (short)0, acc3, false, false);
    }

    // min over the 4 centroids, normalize by 1/D.
    // C/D layout: VGPR r, lanes 0-15 -> row M=r ; lanes 16-31 -> row M=8+r.
    v8f m;
#pragma unroll
    for (int i = 0; i < 8; ++i) {
        const float v01 = fminf(acc0[i], acc1[i]);
        const float v23 = fminf(acc2[i], acc3[i]);
        m[i] = fminf(v01, v23) * (1.0f / 256.0f);
    }

    // Lane 0 holds sums for samples 0..7 (any column), lane 16 for samples 8..15.
    if (row == 0) {
        float* o = out + (long)group * 16 + half * 8;
        *(v4f*)(o + 0) = (v4f){m[0], m[1], m[2], m[3]};
        *(v4f*)(o + 4) = (v4f){m[4], m[5], m[6], m[7]};
    }
}

extern "C" void kernel_launch(void* const* d_in, const int* in_sizes, int n_in,
                              void* d_out, int out_size, void* d_ws, size_t ws_size,
                              hipStream_t stream) {
    const float* codes = (const float*)d_in[0];
    const int*   pred  = (const int*)d_in[1];
    const float* cents = (const float*)d_in[2];
    float*       out   = (float*)d_out;

    const int B      = in_sizes[1];      // 262144 samples
    const int groups = B / 16;           // 16384 wave-groups
    const int blocks = groups / 8;       // 8 waves (256 threads) per block -> 2048 blocks

    _SparseCodebook_kernel<<<blocks, 256, 0, stream>>>(codes, pred, cents, out);
}